// ModernHopfield_66743791780146
// MI455X (gfx1250) — compile-verified
//
#include <hip/hip_runtime.h>
#include <hip/hip_bf16.h>

typedef float v2f __attribute__((ext_vector_type(2)));
typedef float v8f __attribute__((ext_vector_type(8)));

#define BETA 12.0f
#define SIM_SCALE 25.0f

#if defined(__AMDGCN__) && __has_builtin(__builtin_amdgcn_global_load_async_to_lds_b128)
#define USE_ASYNC_LDS 1
typedef int v4i_vs __attribute__((vector_size(16)));
typedef __attribute__((address_space(1))) v4i_vs* g_v4p;   // global (AS1)
typedef __attribute__((address_space(3))) v4i_vs* l_v4p;   // LDS (AS3)
#endif

// ---------------------------------------------------------------------------
// GEMV via V_WMMA_F32_16X16X4_F32:  out[r] = dot(M[r,:], h) * oscale
// Wave = 64 rows = 4 independent 16-row tiles (4 accumulators -> no WMMA RAW
// chain, B fetch amortized 4x). A tile (16x4 f32): lane l<16 = row l, K=0..1
// (b64); lane l>=16 = row l-16, K=2..3 (b64) -- lanes l and l+16 jointly
// stream row l. B (4x16): h[k..k+3] in column N=0 only (lanes 0/16), read
// from LDS-staged copy of h. D column 0 -> lane 0 (rows 0..7 in v[0..7]),
// lane 16 (rows 8..15).
// ---------------------------------------------------------------------------
__global__ void gemv_wmma_kernel(const float* __restrict__ M,
                                 const float* __restrict__ h,
                                 float* __restrict__ out,
                                 int ntiles64, int D, float oscale) {
    __shared__ float hs[512];                    // D <= 512 (reference: 512)
    int t = threadIdx.x;

    // Stage h into LDS once per block (async-to-LDS if available).
    if (t < (D >> 2)) {
#if defined(USE_ASYNC_LDS)
        __builtin_amdgcn_global_load_async_to_lds_b128(
            (g_v4p)(h + 4 * t), (l_v4p)(hs + 4 * t), 0, 0);
#else
        *(float4*)(hs + 4 * t) = *(const float4*)(h + 4 * t);
#endif
    }
#if defined(USE_ASYNC_LDS)
#if __has_builtin(__builtin_amdgcn_s_wait_asynccnt)
    __builtin_amdgcn_s_wait_asynccnt(0);
#else
    asm volatile("s_wait_asynccnt 0x0" ::: "memory");
#endif
#endif
    __syncthreads();

    int lane = t & 31;
    int wave = t >> 5;
    int tile = blockIdx.x * (blockDim.x >> 5) + wave;
    if (tile >= ntiles64) return;                // wave-uniform: EXEC stays full
    int r0 = tile << 6;                          // 64 rows per wave
    int half = lane >> 4;                        // 0: K=0..1, 1: K=2..3
    const float* arow = M + (size_t)(r0 + (lane & 15)) * D + (half << 1);
    const size_t rowblk = (size_t)16 * D;        // 16 rows stride
    const float* hrow = hs + (half << 1);
    bool bl = ((lane & 15) == 0);                // lanes 0/16 carry B column 0

    v8f acc[4] = {};
#pragma unroll 2
    for (int k = 0; k < D; k += 4) {
        v2f b = {0.0f, 0.0f};
        if (bl) b = *(const v2f*)(hrow + k);     // ds_load_b64 from staged h
#pragma unroll
        for (int j = 0; j < 4; ++j) {
            v2f a = *(const v2f*)(arow + (size_t)j * rowblk + k);
            acc[j] = __builtin_amdgcn_wmma_f32_16x16x4_f32(
                /*neg_a=*/false, a, /*neg_b=*/false, b,
                /*c_mod=*/(short)0, acc[j], /*reuse_a=*/false, /*reuse_b=*/false);
        }
    }
    if (bl) {
#pragma unroll
        for (int j = 0; j < 4; ++j) {
            int rb = r0 + (j << 4) + (half << 3);
#pragma unroll
            for (int i = 0; i < 8; ++i) out[rb + i] = acc[j][i] * oscale;
        }
    }
}

// ---------------------------------------------------------------------------
// Online-softmax stage 1: per-block (max, sumexp) over scale*x, fixed order.
// ---------------------------------------------------------------------------
__global__ void softmax_part_kernel(const float* __restrict__ x, int n,
                                    float scale, float* __restrict__ red) {
    __shared__ float smax[256];
    __shared__ float ssum[256];
    int t = threadIdx.x;
    float m = -3.0e38f, s = 0.0f;
    for (int i = blockIdx.x * blockDim.x + t; i < n; i += gridDim.x * blockDim.x) {
        float v = scale * x[i];
        if (v > m) { s = s * __expf(m - v) + 1.0f; m = v; }
        else       { s += __expf(v - m); }
    }
    smax[t] = m; ssum[t] = s;
    __syncthreads();
    for (int off = 128; off > 0; off >>= 1) {
        if (t < off) {
            float m2 = smax[t + off], s2 = ssum[t + off];
            float mm = fmaxf(m, m2);
            s = s * __expf(m - mm) + s2 * __expf(m2 - mm);
            m = mm;
            smax[t] = m; ssum[t] = s;
        }
        __syncthreads();
    }
    if (t == 0) { red[2 * blockIdx.x] = m; red[2 * blockIdx.x + 1] = s; }
}

// Stage 2: combine NB partial (max,sum) pairs -> gred[0]=gmax, gred[1]=gsum.
__global__ void softmax_final_kernel(const float* __restrict__ red, int nb,
                                     float* __restrict__ gred) {
    __shared__ float smax[256];
    __shared__ float ssum[256];
    int t = threadIdx.x;
    float m = -3.0e38f, s = 0.0f;
    for (int i = t; i < nb; i += 256) {
        float m2 = red[2 * i], s2 = red[2 * i + 1];
        float mm = fmaxf(m, m2);
        s = s * __expf(m - mm) + s2 * __expf(m2 - mm);
        m = mm;
    }
    smax[t] = m; ssum[t] = s;
    __syncthreads();
    for (int off = 128; off > 0; off >>= 1) {
        if (t < off) {
            float m2 = smax[t + off], s2 = ssum[t + off];
            float mm = fmaxf(m, m2);
            s = s * __expf(m - mm) + s2 * __expf(m2 - mm);
            m = mm;
            smax[t] = m; ssum[t] = s;
        }
        __syncthreads();
    }
    if (t == 0) { gred[0] = m; gred[1] = s; }
}

__global__ void probs_kernel(const float* __restrict__ x, int n, float scale,
                             const float* __restrict__ gred,
                             float* __restrict__ probs) {
    int i = blockIdx.x * blockDim.x + threadIdx.x;
    if (i < n) {
        float inv = 1.0f / gred[1];
        probs[i] = __expf(scale * x[i] - gred[0]) * inv;
    }
}

// ---------------------------------------------------------------------------
// Pass 1: per-block partial h' = sum_{m in chunk} probs[m] * M[m,:]
// Thread t owns columns 2t, 2t+1 (float2), fully coalesced row streaming.
// ---------------------------------------------------------------------------
__global__ void accum_h_kernel(const float* __restrict__ M,
                               const float* __restrict__ probs,
                               int nrows, int D, int chunk,
                               float* __restrict__ partials) {
    int t = threadIdx.x;
    int c = t << 1;
    int r0 = blockIdx.x * chunk;
    int r1 = min(nrows, r0 + chunk);
    if (c >= D) return;
    const float* p = M + (size_t)r0 * D + c;
    float ax = 0.0f, ay = 0.0f;
    for (int r = r0; r < r1; ++r, p += D) {
        __builtin_prefetch(p + (size_t)8 * D, 0, 0);   // global_prefetch_b8
        float pr = probs[r];
        float2 v = *(const float2*)p;
        ax = fmaf(pr, v.x, ax);
        ay = fmaf(pr, v.y, ay);
    }
    float* o = partials + (size_t)blockIdx.x * D + c;
    o[0] = ax; o[1] = ay;
}

// Deterministic fixed-order reduce of partials; writes hbuf and history row.
__global__ void reduce_h_kernel(const float* __restrict__ partials, int nb,
                                int D, float* __restrict__ hbuf,
                                float* __restrict__ hist) {
    int d = blockIdx.x * blockDim.x + threadIdx.x;
    if (d >= D) return;
    float s = 0.0f;
    for (int b = 0; b < nb; ++b) s += partials[(size_t)b * D + d];
    hbuf[d] = s;
    hist[d] = s;
}

extern "C" void kernel_launch(void* const* d_in, const int* in_sizes, int n_in,
                              void* d_out, int out_size, void* d_ws, size_t ws_size,
                              hipStream_t stream) {
    const float* M     = (const float*)d_in[0];
    const float* state = (const float*)d_in[1];

    const int D     = in_sizes[1];                       // 512
    const int nrows = in_sizes[0] / D;                   // 200000
    const int steps = (out_size - D) / (nrows + D);      // 30

    float* hist = (float*)d_out;                         // [steps+1, D]
    float* sims = (float*)d_out + (size_t)(steps + 1) * D; // [steps, nrows]

    // Workspace carve-up (floats)
    const int NB1 = 500;   // pass-1 partial blocks
    const int NB2 = 256;   // softmax partial blocks
    float* ws       = (float*)d_ws;
    float* logits0  = ws;                 ws += nrows;
    float* probs    = ws;                 ws += nrows;
    float* hbuf     = ws;                 ws += D;
    float* partials = ws;                 ws += (size_t)NB1 * D;
    float* red      = ws;                 ws += 2 * NB2;
    float* gred     = ws;                 ws += 2;

    const int ntiles64  = nrows / 64;                    // 3125 (nrows % 64 == 0)
    const int wmma_blk  = 256;                           // 8 waves/block
    const int wmma_grid = (ntiles64 + 7) / 8;
    const int chunk     = (nrows + NB1 - 1) / NB1;

    // history[0] = initial state
    (void)hipMemcpyAsync(hist, state, (size_t)D * sizeof(float),
                         hipMemcpyDeviceToDevice, stream);

    // logits0 = M @ h0
    gemv_wmma_kernel<<<wmma_grid, wmma_blk, 0, stream>>>(M, state, logits0,
                                                         ntiles64, D, 1.0f);

    for (int t = 0; t < steps; ++t) {
        // logits for this step: step 0 uses logits0; later steps reuse sims*25
        const float* xsrc  = (t == 0) ? logits0 : (sims + (size_t)(t - 1) * nrows);
        const float  xscal = (t == 0) ? BETA : BETA * SIM_SCALE;

        softmax_part_kernel<<<NB2, 256, 0, stream>>>(xsrc, nrows, xscal, red);
        softmax_final_kernel<<<1, 256, 0, stream>>>(red, NB2, gred);
        probs_kernel<<<(nrows + 255) / 256, 256, 0, stream>>>(xsrc, nrows,
                                                              xscal, gred, probs);

        // h' = M^T @ probs  (partials, then deterministic reduce -> hbuf + history)
        accum_h_kernel<<<NB1, 256, 0, stream>>>(M, probs, nrows, D, chunk, partials);
        reduce_h_kernel<<<(D + 255) / 256, 256, 0, stream>>>(
            partials, NB1, D, hbuf, hist + (size_t)(t + 1) * D);

        // sims_t = (M @ h') / 25   (doubles as next step's logits, rescaled)
        gemv_wmma_kernel<<<wmma_grid, wmma_blk, 0, stream>>>(
            M, hbuf, sims + (size_t)t * nrows, ntiles64, D, 1.0f / SIM_SCALE);
    }
}